// GNNModel_1322849927837
// MI455X (gfx1250) — compile-verified
//
#include <hip/hip_runtime.h>
#include <hip/hip_bf16.h>
#include <math.h>
#include <stdint.h>

typedef __attribute__((ext_vector_type(2))) float v2f;
typedef __attribute__((ext_vector_type(8))) float v8f;

#define HID    128
#define NCLS   10
#define NGRAPH 64

#if __has_builtin(__builtin_amdgcn_wmma_f32_16x16x4_f32)
#define HAVE_WMMA_F32 1
#else
#define HAVE_WMMA_F32 0
#endif

// ---------------- degree / dinv ----------------
__global__ void k_deg_init(float* deg, int n) {
    int i = blockIdx.x * blockDim.x + threadIdx.x;
    if (i < n) deg[i] = 1.0f;  // self-loop contributes 1 to every node's degree
}

__global__ void k_deg_edges(const int* __restrict__ dst, float* deg, int nE) {
    int e = blockIdx.x * blockDim.x + threadIdx.x;
    if (e < nE) atomicAdd(&deg[dst[e]], 1.0f);
}

__global__ void k_deg_to_dinv(float* d, int n) {
    int i = blockIdx.x * blockDim.x + threadIdx.x;
    if (i < n) d[i] = rsqrtf(d[i]);  // deg >= 1 always (self loops)
}

// ---------------- GEMM: H[M,128] = X[M,128] @ W[128,128] (fp32 WMMA) ----------------
// Block = 256 threads (8 waves). W (64 KB) is staged once per block into LDS via
// async global->LDS copy (GLOBAL_LOAD_ASYNC_TO_LDS_B128, tracked by ASYNCcnt);
// each wave then computes a 16x128 output stripe with 8 chained
// V_WMMA_F32_16X16X4_F32 accumulators fed from ds_load.
__global__ __launch_bounds__(256) void k_gemm128(const float* __restrict__ X,
                                                 const float* __restrict__ W,
                                                 float* __restrict__ H,
                                                 int rowTiles) {
    __shared__ float sW[HID * HID];  // 64 KB of the 320 KB/WGP LDS

    // ---- stage W into LDS (all threads participate) ----
    {
        const float4* Wv = (const float4*)W;
        float4* sWv = (float4*)sW;
#if defined(__gfx1250__)
        for (int i = threadIdx.x; i < (HID * HID) / 4; i += 256) {
            uint32_t ldsa = (uint32_t)(uintptr_t)(sWv + i);  // low 32 bits = LDS offset
            uint64_t ga   = (uint64_t)(uintptr_t)(Wv + i);
            asm volatile("global_load_async_to_lds_b128 %0, %1, off"
                         :: "v"(ldsa), "v"(ga)
                         : "memory");
        }
        asm volatile("s_wait_asynccnt 0" ::: "memory");
#else
        for (int i = threadIdx.x; i < (HID * HID) / 4; i += 256) {
            sWv[i] = Wv[i];  // global_load_b128 + ds_store_b128
        }
#endif
    }
    __syncthreads();

    int wave = blockIdx.x * (blockDim.x >> 5) + (threadIdx.x >> 5);
    if (wave < rowTiles) {  // wave-uniform: EXEC stays all-ones for WMMA
        int lane = threadIdx.x & 31;
        int half = lane >> 4;   // 0: K pair {0,1}, 1: K pair {2,3}
        int l16  = lane & 15;

#if HAVE_WMMA_F32
        const float* xrow = X + (size_t)(wave * 16 + l16) * HID;  // A: M = l16
        v8f acc[8] = {};
        for (int k = 0; k < HID; k += 4) {
            int k0 = k + 2 * half;
            v2f a;
            a.x = xrow[k0];
            a.y = xrow[k0 + 1];
#pragma unroll
            for (int t = 0; t < 8; ++t) {
                int col = t * 16 + l16;  // B: N = l16
                v2f b;
                b.x = sW[k0 * HID + col];
                b.y = sW[(k0 + 1) * HID + col];
                acc[t] = __builtin_amdgcn_wmma_f32_16x16x4_f32(
                    false, a, false, b, (short)0, acc[t], false, false);
            }
        }
        // C/D layout: VGPR v -> row (v + 8*half), lane%16 -> col
        int rbase = wave * 16 + 8 * half;
#pragma unroll
        for (int t = 0; t < 8; ++t) {
#pragma unroll
            for (int v = 0; v < 8; ++v) {
                H[(size_t)(rbase + v) * HID + t * 16 + l16] = acc[t][v];
            }
        }
#else
        // scalar fp32 fallback (correctness only)
        for (int r = 0; r < 16; ++r) {
            int rr = wave * 16 + r;
            for (int c = lane; c < HID; c += 32) {
                float s = 0.0f;
                for (int k = 0; k < HID; ++k) s += X[(size_t)rr * HID + k] * sW[k * HID + c];
                H[(size_t)rr * HID + c] = s;
            }
        }
#endif
    }
}

// ---------------- agg init: agg[i] = h[i] * dinv[i]^2 (self-loop message) ----------------
__global__ void k_agg_init(const float* __restrict__ h, const float* __restrict__ dinv,
                           float* __restrict__ agg, int n) {
    int idx = blockIdx.x * blockDim.x + threadIdx.x;  // n*32 threads, float4 each
    if (idx >= n * 32) return;
    int node = idx >> 5;
    int q = idx & 31;
    float d = dinv[node];
    float nn = d * d;
    float4 v = reinterpret_cast<const float4*>(h + (size_t)node * HID)[q];
    float4 o = make_float4(v.x * nn, v.y * nn, v.z * nn, v.w * nn);
    reinterpret_cast<float4*>(agg + (size_t)node * HID)[q] = o;
}

// ---------------- edge scatter: agg[dst] += h[src] * dinv[src]*dinv[dst] ----------------
// one wave32 per edge; 4 features per lane (float4 gather + 4 f32 atomics, L2-resident)
__global__ __launch_bounds__(256) void k_edge_agg(const int* __restrict__ src,
                                                  const int* __restrict__ dst,
                                                  const float* __restrict__ h,
                                                  const float* __restrict__ dinv,
                                                  float* __restrict__ agg, int nE) {
    int wave = blockIdx.x * (blockDim.x >> 5) + (threadIdx.x >> 5);
    if (wave >= nE) return;
    int lane = threadIdx.x & 31;
    int s = src[wave];
    int d = dst[wave];
    float norm = dinv[s] * dinv[d];
    float4 v = reinterpret_cast<const float4*>(h + (size_t)s * HID)[lane];
    float* o = agg + (size_t)d * HID + lane * 4;
    atomicAdd(o + 0, v.x * norm);
    atomicAdd(o + 1, v.y * norm);
    atomicAdd(o + 2, v.z * norm);
    atomicAdd(o + 3, v.w * norm);
}

// ---------------- bias + ReLU (in place) ----------------
__global__ void k_bias_relu(float* __restrict__ agg, const float* __restrict__ b, int total) {
    int i = blockIdx.x * blockDim.x + threadIdx.x;
    if (i < total) agg[i] = fmaxf(agg[i] + b[i & (HID - 1)], 0.0f);
}

// ---------------- pooling ----------------
__global__ void k_pool_zero(float* gsum, float* cnt) {
    int i = blockIdx.x * blockDim.x + threadIdx.x;
    if (i < NGRAPH * HID) gsum[i] = 0.0f;
    if (i < NGRAPH) cnt[i] = 0.0f;
}

__global__ __launch_bounds__(256) void k_pool(const float* __restrict__ h,
                                              const int* __restrict__ batch,
                                              float* __restrict__ gsum, float* __restrict__ cnt,
                                              int n) {
    int wave = blockIdx.x * (blockDim.x >> 5) + (threadIdx.x >> 5);
    if (wave >= n) return;
    int lane = threadIdx.x & 31;
    int g = batch[wave];
    float4 v = reinterpret_cast<const float4*>(h + (size_t)wave * HID)[lane];
    float* o = gsum + (size_t)g * HID + lane * 4;
    atomicAdd(o + 0, v.x);
    atomicAdd(o + 1, v.y);
    atomicAdd(o + 2, v.z);
    atomicAdd(o + 3, v.w);
    if (lane == 0) atomicAdd(&cnt[g], 1.0f);
}

// ---------------- MLP head ----------------
__global__ void k_head1(const float* __restrict__ gsum, const float* __restrict__ cnt,
                        const float* __restrict__ Wc1, const float* __restrict__ bc1,
                        float* __restrict__ z) {
    int i = blockIdx.x * blockDim.x + threadIdx.x;
    if (i >= NGRAPH * HID) return;
    int g = i / HID, j = i % HID;
    float inv = 1.0f / fmaxf(cnt[g], 1.0f);
    float s = bc1[j];
    for (int k = 0; k < HID; ++k) s += (gsum[(size_t)g * HID + k] * inv) * Wc1[(size_t)k * HID + j];
    z[i] = fmaxf(s, 0.0f);
}

__global__ void k_head2(const float* __restrict__ z, const float* __restrict__ Wc2,
                        const float* __restrict__ bc2, float* __restrict__ out) {
    int i = blockIdx.x * blockDim.x + threadIdx.x;
    if (i >= NGRAPH * NCLS) return;
    int g = i / NCLS, c = i % NCLS;
    float s = bc2[c];
    for (int k = 0; k < HID; ++k) s += z[(size_t)g * HID + k] * Wc2[(size_t)k * NCLS + c];
    out[i] = s;
}

extern "C" void kernel_launch(void* const* d_in, const int* in_sizes, int n_in,
                              void* d_out, int out_size, void* d_ws, size_t ws_size,
                              hipStream_t stream) {
    const float* x    = (const float*)d_in[0];
    const int*   ei   = (const int*)d_in[1];
    const int*   bat  = (const int*)d_in[2];
    const float* W1   = (const float*)d_in[3];
    const float* b1   = (const float*)d_in[4];
    const float* W2   = (const float*)d_in[5];
    const float* b2   = (const float*)d_in[6];
    const float* Wc1  = (const float*)d_in[7];
    const float* bc1  = (const float*)d_in[8];
    const float* Wc2  = (const float*)d_in[9];
    const float* bc2  = (const float*)d_in[10];
    float* out = (float*)d_out;

    int nNodes = in_sizes[0] / HID;   // 50000
    int nEdges = in_sizes[1] / 2;     // 1.6M
    const int* esrc = ei;
    const int* edst = ei + nEdges;

    // workspace carve-up
    float* h    = (float*)d_ws;                      // nNodes*128
    float* agg  = h + (size_t)nNodes * HID;          // nNodes*128
    float* dinv = agg + (size_t)nNodes * HID;        // nNodes
    float* gsum = dinv + nNodes;                     // 64*128
    float* cnt  = gsum + NGRAPH * HID;               // 64
    float* z    = cnt + NGRAPH;                      // 64*128

    const int T = 256;
    int rowTiles = nNodes / 16;                      // 3125
    dim3 gemmGrid((rowTiles + 7) / 8);
    dim3 edgeGrid((nEdges + 7) / 8);
    dim3 nodeWaveGrid((nNodes + 7) / 8);

    // degrees -> dinv
    k_deg_init<<<(nNodes + T - 1) / T, T, 0, stream>>>(dinv, nNodes);
    k_deg_edges<<<(nEdges + T - 1) / T, T, 0, stream>>>(edst, dinv, nEdges);
    k_deg_to_dinv<<<(nNodes + T - 1) / T, T, 0, stream>>>(dinv, nNodes);

    // ---- layer 1 ----
    k_gemm128<<<gemmGrid, T, 0, stream>>>(x, W1, h, rowTiles);
    k_agg_init<<<(nNodes * 32 + T - 1) / T, T, 0, stream>>>(h, dinv, agg, nNodes);
    k_edge_agg<<<edgeGrid, T, 0, stream>>>(esrc, edst, h, dinv, agg, nEdges);
    k_bias_relu<<<(nNodes * HID + T - 1) / T, T, 0, stream>>>(agg, b1, nNodes * HID);

    // ---- layer 2 (agg now holds layer-1 activations; reuse h, then overwrite agg) ----
    k_gemm128<<<gemmGrid, T, 0, stream>>>(agg, W2, h, rowTiles);
    k_agg_init<<<(nNodes * 32 + T - 1) / T, T, 0, stream>>>(h, dinv, agg, nNodes);
    k_edge_agg<<<edgeGrid, T, 0, stream>>>(esrc, edst, h, dinv, agg, nEdges);
    k_bias_relu<<<(nNodes * HID + T - 1) / T, T, 0, stream>>>(agg, b2, nNodes * HID);

    // ---- pooling + head ----
    k_pool_zero<<<(NGRAPH * HID + T - 1) / T, T, 0, stream>>>(gsum, cnt);
    k_pool<<<nodeWaveGrid, T, 0, stream>>>(agg, bat, gsum, cnt, nNodes);
    k_head1<<<(NGRAPH * HID + T - 1) / T, T, 0, stream>>>(gsum, cnt, Wc1, bc1, z);
    k_head2<<<(NGRAPH * NCLS + T - 1) / T, T, 0, stream>>>(z, Wc2, bc2, out);
}